// ISDALoss_69063074120048
// MI455X (gfx1250) — compile-verified
//
#include <hip/hip_runtime.h>
#include <hip/hip_bf16.h>

typedef float v2f __attribute__((ext_vector_type(2)));
typedef float v8f __attribute__((ext_vector_type(8)));

#define A_DIM 512
#define HW    16384
#define N_IMG 4
#define C_CLS 19
#define C_PAD 32
#define M_PIX 65536
#define CHUNK 1024

// d_ws float layout:
//   [0,16384)      sum_f   [32][512]
//   [16384,32768)  sum_sq  [32][512]
//   [32768,32800)  counts  [32]
//   [32800,33161)  S       [19*19]
#define WS_SUM    0
#define WS_SUMSQ  16384
#define WS_CNT    32768
#define WS_S      32800
#define WS_ZERO_N 32800

__global__ void __launch_bounds__(256) zero_kernel(float* ws) {
    int i = blockIdx.x * 256 + threadIdx.x;
    if (i < WS_ZERO_N) ws[i] = 0.0f;
}

// -------- label histogram (counts per class) via LDS atomics --------
__global__ void __launch_bounds__(256) count_kernel(const int* __restrict__ labels,
                                                    float* __restrict__ counts) {
    __shared__ unsigned hist[C_PAD];
    const int tid = threadIdx.x;
    if (tid < C_PAD) hist[tid] = 0u;
    __syncthreads();
    for (int i = blockIdx.x * 256 + tid; i < M_PIX; i += gridDim.x * 256) {
        int lb = labels[i];
        if ((unsigned)lb < (unsigned)C_PAD) atomicAdd(&hist[lb], 1u);
    }
    __syncthreads();
    if (tid < C_PAD && hist[tid] != 0u) atomicAdd(&counts[tid], (float)hist[tid]);
}

// -------- phase 1: segment sums of f and f^2 via f32 WMMA --------
// D = onehot(16x4) x F(4x16) + C, accumulated over 256 K-groups per block.
// grid: x = 64 pixel chunks of 1024, y = 8 groups of 64 features; 4 waves/block,
// each wave owns a 16-feature sub-tile.
__global__ void __launch_bounds__(128) phase1_wmma(const float* __restrict__ feats,
                                                   const int*   __restrict__ labels,
                                                   float* __restrict__ sum_g,
                                                   float* __restrict__ sumsq_g) {
    __shared__ int labs[CHUNK];
    const int chunk  = blockIdx.x;     // 0..63
    const int aGroup = blockIdx.y;     // 0..7
    const int tid  = threadIdx.x;
    const int lane = tid & 31;
    const int wave = tid >> 5;

    const int img   = chunk >> 4;            // 16 chunks per image
    const int pbase = (chunk & 15) * CHUNK;  // pixel base within image
    const int gbase = chunk * CHUNK;         // flat pixel base (labels)

    for (int i = tid; i < CHUNK; i += 128) labs[i] = labels[gbase + i];
    __syncthreads();

    const int row = lane & 15;        // class row (A-tile) == feature col (B/D)
    const int kp  = (lane >> 4) * 2;  // this half-wave's K pair {0,1} or {2,3}
    const int a   = aGroup * 64 + wave * 16 + row;
    const float* fptr = feats + ((size_t)(img * A_DIM + a) * HW + pbase);

    v8f acc0 = {}; v8f acc1 = {}; v8f acc2 = {}; v8f acc3 = {};

    for (int kg = 0; kg < CHUNK / 4; ++kg) {
        const int p0 = kg * 4;
        if ((kg & 7) == 0) __builtin_prefetch(fptr + p0 + 64, 0, 0); // global_prefetch_b8

        const int2   lb = *(const int2*)(&labs[p0 + kp]);
        const float2 fv = *(const float2*)(fptr + p0 + kp);

        v2f b;   b[0] = fv.x;         b[1] = fv.y;
        v2f bsq; bsq[0] = fv.x * fv.x; bsq[1] = fv.y * fv.y;
        v2f a0, a1;
        a0[0] = (lb.x == row)      ? 1.0f : 0.0f;
        a0[1] = (lb.y == row)      ? 1.0f : 0.0f;
        a1[0] = (lb.x == row + 16) ? 1.0f : 0.0f;
        a1[1] = (lb.y == row + 16) ? 1.0f : 0.0f;

        acc0 = __builtin_amdgcn_wmma_f32_16x16x4_f32(false, a0, false, b,   (short)0, acc0, false, false);
        acc1 = __builtin_amdgcn_wmma_f32_16x16x4_f32(false, a1, false, b,   (short)0, acc1, false, false);
        acc2 = __builtin_amdgcn_wmma_f32_16x16x4_f32(false, a0, false, bsq, (short)0, acc2, false, false);
        acc3 = __builtin_amdgcn_wmma_f32_16x16x4_f32(false, a1, false, bsq, (short)0, acc3, false, false);
    }

    // D layout: col n = lane&15, row m = vgpr + 8*(lane>=16)
    const int mbase = (lane >> 4) * 8;
    const int acol  = a;  // == aGroup*64 + wave*16 + (lane&15)
#pragma unroll
    for (int v = 0; v < 8; ++v) {
        const int m = mbase + v;
        atomicAdd(&sum_g  [(size_t)m        * A_DIM + acol], acc0[v]);
        atomicAdd(&sum_g  [(size_t)(m + 16) * A_DIM + acol], acc1[v]);
        atomicAdd(&sumsq_g[(size_t)m        * A_DIM + acol], acc2[v]);
        atomicAdd(&sumsq_g[(size_t)(m + 16) * A_DIM + acol], acc3[v]);
    }
}

// -------- phase 2+3: cov (in LDS) then S[k,c] table --------
__global__ void __launch_bounds__(512) stats_kernel(const float* __restrict__ sum_g,
                                                    const float* __restrict__ sumsq_g,
                                                    const float* __restrict__ counts,
                                                    const float* __restrict__ Cov_in,
                                                    const float* __restrict__ Ave_in,
                                                    const float* __restrict__ Amount_in,
                                                    const float* __restrict__ weight,
                                                    const float* __restrict__ ratio_p,
                                                    float* __restrict__ S_out) {
    __shared__ float cov_s[C_CLS * A_DIM];  // 38 KB
    const int tid = threadIdx.x;
    for (int idx = tid; idx < C_CLS * A_DIM; idx += 512) {
        const int c = idx >> 9;            // / A_DIM
        const float cnt = counts[c];
        const float cs  = (cnt == 0.0f) ? 1.0f : cnt;
        const float ave = sum_g[idx] / cs;
        float var = sumsq_g[idx] / cs - ave * ave;
        var = fmaxf(var, 0.0f);
        const float den = cnt + Amount_in[c];
        const float w   = (den == 0.0f) ? 0.0f : cnt / den;
        const float d   = Ave_in[idx] - ave;
        cov_s[idx] = Cov_in[idx] * (1.0f - w) + var * w + w * (1.0f - w) * d * d;
    }
    __syncthreads();
    const float ratio = ratio_p[0];
    for (int p = tid; p < C_CLS * C_CLS; p += 512) {
        const int k = p / C_CLS, c = p % C_CLS;
        const float* wc = weight + (size_t)c * A_DIM;
        const float* wk = weight + (size_t)k * A_DIM;
        const float* cv = cov_s + k * A_DIM;
        float s = 0.0f;
        for (int a2 = 0; a2 < A_DIM; ++a2) {
            const float d = wc[a2] - wk[a2];
            s = fmaf(d * d, cv[a2], s);
        }
        S_out[p] = ratio * s;
    }
}

// -------- phase 4: out[n,c,h,w] = y + 0.5 * S[lab, c] --------
__global__ void __launch_bounds__(256) out_kernel(const float* __restrict__ y,
                                                  const int*   __restrict__ labels,
                                                  const float* __restrict__ S,
                                                  float* __restrict__ out) {
    __shared__ float s_s[C_CLS * C_CLS];
    const int tid = threadIdx.x;
    for (int i = tid; i < C_CLS * C_CLS; i += 256) s_s[i] = S[i];
    __syncthreads();

    const int idx4 = blockIdx.x * 256 + tid;            // float4 index
    const int total4 = N_IMG * C_CLS * HW / 4;          // 311296
    if (idx4 >= total4) return;
    const int idx = idx4 * 4;
    const int p  = idx & (HW - 1);
    const int nc = idx / HW;
    const int c  = nc % C_CLS;
    const int n  = nc / C_CLS;

    const int4   lb = *(const int4*)(labels + (size_t)n * HW + p);
    const float4 yv = *(const float4*)(y + idx);
    float4 o;
    o.x = yv.x + (((unsigned)lb.x < (unsigned)C_CLS) ? 0.5f * s_s[lb.x * C_CLS + c] : 0.0f);
    o.y = yv.y + (((unsigned)lb.y < (unsigned)C_CLS) ? 0.5f * s_s[lb.y * C_CLS + c] : 0.0f);
    o.z = yv.z + (((unsigned)lb.z < (unsigned)C_CLS) ? 0.5f * s_s[lb.z * C_CLS + c] : 0.0f);
    o.w = yv.w + (((unsigned)lb.w < (unsigned)C_CLS) ? 0.5f * s_s[lb.w * C_CLS + c] : 0.0f);
    *(float4*)(out + idx) = o;
}

extern "C" void kernel_launch(void* const* d_in, const int* in_sizes, int n_in,
                              void* d_out, int out_size, void* d_ws, size_t ws_size,
                              hipStream_t stream) {
    const float* feats  = (const float*)d_in[0];   // [4,512,128,128]
    const float* y      = (const float*)d_in[1];   // [4,19,128,128]
    const int*   labels = (const int*)  d_in[2];   // [4,1,128,128]
    const float* weight = (const float*)d_in[3];   // [19,512]
    const float* Cov    = (const float*)d_in[4];   // [19,512]
    const float* Ave    = (const float*)d_in[5];   // [19,512]
    const float* Amount = (const float*)d_in[6];   // [19]
    const float* ratio  = (const float*)d_in[7];   // scalar

    float* ws      = (float*)d_ws;
    float* sum_g   = ws + WS_SUM;
    float* sumsq_g = ws + WS_SUMSQ;
    float* counts  = ws + WS_CNT;
    float* S       = ws + WS_S;
    float* out     = (float*)d_out;

    zero_kernel<<<(WS_ZERO_N + 255) / 256, 256, 0, stream>>>(ws);
    count_kernel<<<64, 256, 0, stream>>>(labels, counts);
    dim3 g1(64, 8);
    phase1_wmma<<<g1, 128, 0, stream>>>(feats, labels, sum_g, sumsq_g);
    stats_kernel<<<1, 512, 0, stream>>>(sum_g, sumsq_g, counts, Cov, Ave, Amount,
                                        weight, ratio, S);
    out_kernel<<<1216, 256, 0, stream>>>(y, labels, S, out);
}